// TSDFFusion_3513283248983
// MI455X (gfx1250) — compile-verified
//
#include <hip/hip_runtime.h>
#include <cstdint>

typedef float f32x4 __attribute__((ext_vector_type(4)));
typedef int   v4i   __attribute__((ext_vector_type(4)));

namespace {
constexpr int   kGrid    = 256;
constexpr int   kNVox    = kGrid * kGrid * kGrid;     // 16,777,216
constexpr int   kW       = 640;
constexpr int   kH       = 480;
constexpr float kTrunc   = 0.1f;
constexpr float kMaxW    = 100.0f;
constexpr float kLo      = -1.9921875f;               // BOUND_MIN + VOXEL/2 (exact fp32)
constexpr float kStep    = 0.015625f;                 // 4/256 (exact fp32)

constexpr int kBlock       = 256;                     // 8 waves (wave32)
constexpr int kVpt         = 4;                       // voxels/thread, contiguous in z
constexpr int kTileVox     = kBlock * kVpt;           // 1024 voxels / tile
constexpr int kNTiles      = kNVox / kTileVox;        // 16384
constexpr int kNBlk        = 2048;                    // persistent blocks
constexpr int kTilesPerBlk = kNTiles / kNBlk;         // 8 tiles per block
} // namespace

// ---- CDNA5 async global->LDS copy path (guarded: falls back to plain loads) ----
#if defined(__gfx1250__) && __has_builtin(__builtin_amdgcn_global_load_async_to_lds_b128)
#define HAVE_ASYNC_LDS 1
#else
#define HAVE_ASYNC_LDS 0
#endif

#if HAVE_ASYNC_LDS
__device__ __forceinline__ void async_copy_b128(const float* g, float* l) {
  // Builtin prototype (from clang diagnostic): param0 = int4 AS1*, non-const.
  // cpol = 1 -> TH=NT (ISA load temporal hints: 0=RT,1=NT): the volume stream is
  // touch-once, keep it from evicting the depth map out of L2.
  __builtin_amdgcn_global_load_async_to_lds_b128(
      (__attribute__((address_space(1))) v4i*)g,
      (__attribute__((address_space(3))) v4i*)l,
      /*imm offset*/ 0, /*cpol NT*/ 1);
}
__device__ __forceinline__ void wait_asynccnt_le2() {
#if __has_builtin(__builtin_amdgcn_s_wait_asynccnt)
  __builtin_amdgcn_s_wait_asynccnt(2);
#else
  asm volatile("s_wait_asynccnt 2" ::: "memory");
#endif
}
__device__ __forceinline__ void wait_asynccnt_le0() {
#if __has_builtin(__builtin_amdgcn_s_wait_asynccnt)
  __builtin_amdgcn_s_wait_asynccnt(0);
#else
  asm volatile("s_wait_asynccnt 0" ::: "memory");
#endif
}
#endif

// ------------------------------------------------------------------
// Prep: world2cam = inv(pose); Q[0..2] = K @ world2cam[0:3,:]; Q[3] = world2cam[2,:]
// ------------------------------------------------------------------
__global__ void tsdf_prep_kernel(const float* __restrict__ pose,
                                 const float* __restrict__ intr,
                                 float* __restrict__ Q) {
  if (threadIdx.x != 0 || blockIdx.x != 0) return;
  float m[16], inv[16], w[16];
#pragma unroll
  for (int i = 0; i < 16; ++i) m[i] = pose[i];

  inv[0]  =  m[5]*m[10]*m[15] - m[5]*m[11]*m[14] - m[9]*m[6]*m[15] + m[9]*m[7]*m[14] + m[13]*m[6]*m[11] - m[13]*m[7]*m[10];
  inv[4]  = -m[4]*m[10]*m[15] + m[4]*m[11]*m[14] + m[8]*m[6]*m[15] - m[8]*m[7]*m[14] - m[12]*m[6]*m[11] + m[12]*m[7]*m[10];
  inv[8]  =  m[4]*m[9]*m[15]  - m[4]*m[11]*m[13] - m[8]*m[5]*m[15] + m[8]*m[7]*m[13] + m[12]*m[5]*m[11] - m[12]*m[7]*m[9];
  inv[12] = -m[4]*m[9]*m[14]  + m[4]*m[10]*m[13] + m[8]*m[5]*m[14] - m[8]*m[6]*m[13] - m[12]*m[5]*m[10] + m[12]*m[6]*m[9];
  inv[1]  = -m[1]*m[10]*m[15] + m[1]*m[11]*m[14] + m[9]*m[2]*m[15] - m[9]*m[3]*m[14] - m[13]*m[2]*m[11] + m[13]*m[3]*m[10];
  inv[5]  =  m[0]*m[10]*m[15] - m[0]*m[11]*m[14] - m[8]*m[2]*m[15] + m[8]*m[3]*m[14] + m[12]*m[2]*m[11] - m[12]*m[3]*m[10];
  inv[9]  = -m[0]*m[9]*m[15]  + m[0]*m[11]*m[13] + m[8]*m[1]*m[15] - m[8]*m[3]*m[13] - m[12]*m[1]*m[11] + m[12]*m[3]*m[9];
  inv[13] =  m[0]*m[9]*m[14]  - m[0]*m[10]*m[13] - m[8]*m[1]*m[14] + m[8]*m[2]*m[13] + m[12]*m[1]*m[10] - m[12]*m[2]*m[9];
  inv[2]  =  m[1]*m[6]*m[15]  - m[1]*m[7]*m[14]  - m[5]*m[2]*m[15] + m[5]*m[3]*m[14] + m[13]*m[2]*m[7]  - m[13]*m[3]*m[6];
  inv[6]  = -m[0]*m[6]*m[15]  + m[0]*m[7]*m[14]  + m[4]*m[2]*m[15] - m[4]*m[3]*m[14] - m[12]*m[2]*m[7]  + m[12]*m[3]*m[6];
  inv[10] =  m[0]*m[5]*m[15]  - m[0]*m[7]*m[13]  - m[4]*m[1]*m[15] + m[4]*m[3]*m[13] + m[12]*m[1]*m[7]  - m[12]*m[3]*m[5];
  inv[14] = -m[0]*m[5]*m[14]  + m[0]*m[6]*m[13]  + m[4]*m[1]*m[14] - m[4]*m[2]*m[13] - m[12]*m[1]*m[6]  + m[12]*m[2]*m[5];
  inv[3]  = -m[1]*m[6]*m[11]  + m[1]*m[7]*m[10]  + m[5]*m[2]*m[11] - m[5]*m[3]*m[10] - m[9]*m[2]*m[7]   + m[9]*m[3]*m[6];
  inv[7]  =  m[0]*m[6]*m[11]  - m[0]*m[7]*m[10]  - m[4]*m[2]*m[11] + m[4]*m[3]*m[10] + m[8]*m[2]*m[7]   - m[8]*m[3]*m[6];
  inv[11] = -m[0]*m[5]*m[11]  + m[0]*m[7]*m[9]   + m[4]*m[1]*m[11] - m[4]*m[3]*m[9]  - m[8]*m[1]*m[7]   + m[8]*m[3]*m[5];
  inv[15] =  m[0]*m[5]*m[10]  - m[0]*m[6]*m[9]   - m[4]*m[1]*m[10] + m[4]*m[2]*m[9]  + m[8]*m[1]*m[6]   - m[8]*m[2]*m[5];

  const float det  = m[0]*inv[0] + m[1]*inv[4] + m[2]*inv[8] + m[3]*inv[12];
  const float rdet = 1.0f / det;
#pragma unroll
  for (int i = 0; i < 16; ++i) w[i] = inv[i] * rdet;

  // Q rows 0..2 = K(3x3) @ W(rows 0..2), Q row 3 = W row 2 (voxel depth)
#pragma unroll
  for (int r = 0; r < 3; ++r)
#pragma unroll
    for (int c = 0; c < 4; ++c)
      Q[r*4 + c] = intr[r*3 + 0]*w[0*4 + c] + intr[r*3 + 1]*w[1*4 + c] + intr[r*3 + 2]*w[2*4 + c];
#pragma unroll
  for (int c = 0; c < 4; ++c) Q[12 + c] = w[8 + c];
}

// ------------------------------------------------------------------
// Main fusion kernel: persistent blocks, async double-buffered stream
// ------------------------------------------------------------------
__global__ void __launch_bounds__(kBlock)
tsdf_fuse_kernel(const float* __restrict__ depth,
                 const float* __restrict__ Qm,
                 const float* __restrict__ tsdf_in,
                 const float* __restrict__ wght_in,
                 float* __restrict__ tsdf_out,
                 float* __restrict__ wght_out) {
#if HAVE_ASYNC_LDS
  __shared__ float lds_t[2][kTileVox];
  __shared__ float lds_w[2][kTileVox];
#endif
  const int t = threadIdx.x;
  const int b = blockIdx.x;

  // Combined projection matrix (uniform; 64B, hits L2)
  const f32x4 q0 = *(const f32x4*)(Qm + 0);
  const f32x4 q1 = *(const f32x4*)(Qm + 4);
  const f32x4 q2 = *(const f32x4*)(Qm + 8);
  const f32x4 q3 = *(const f32x4*)(Qm + 12);

#if HAVE_ASYNC_LDS
  { // prologue: prefetch this block's first tile
    const int g0 = b * kTileVox + t * kVpt;
    async_copy_b128(tsdf_in + g0, &lds_t[0][t * kVpt]);
    async_copy_b128(wght_in + g0, &lds_w[0][t * kVpt]);
  }
#endif

  for (int i = 0; i < kTilesPerBlk; ++i) {
    const int tile  = b + i * kNBlk;           // strided persistent schedule
    const int gbase = tile * kTileVox + t * kVpt;

    f32x4 tv, wv;
#if HAVE_ASYNC_LDS
    const int buf = i & 1;
    if (i + 1 < kTilesPerBlk) {
      const int gn = (tile + kNBlk) * kTileVox + t * kVpt;
      async_copy_b128(tsdf_in + gn, &lds_t[buf ^ 1][t * kVpt]);
      async_copy_b128(wght_in + gn, &lds_w[buf ^ 1][t * kVpt]);
      wait_asynccnt_le2();   // current tile's 2 copies done (in-order completion)
    } else {
      wait_asynccnt_le0();
    }
    tv = *(const f32x4*)&lds_t[buf][t * kVpt]; // ds_load_b128, own wave's data
    wv = *(const f32x4*)&lds_w[buf][t * kVpt];
#else
    tv = __builtin_nontemporal_load((const f32x4*)(tsdf_in + gbase));
    wv = __builtin_nontemporal_load((const f32x4*)(wght_in + gbase));
#endif

    // voxel coords from linear index (bit-exact vs linspace: multiples of 1/128)
    const int x  = gbase >> 16;
    const int y  = (gbase >> 8) & 255;
    const int z0 = gbase & 255;
    const float vx = kLo + (float)x * kStep;
    const float vy = kLo + (float)y * kStep;
    const float vz0 = kLo + (float)z0 * kStep;

    // hoist x/y terms of the 4x4 projection (z varies across the 4 voxels)
    const float b0 = q0.x * vx + q0.y * vy + q0.w;
    const float b1 = q1.x * vx + q1.y * vy + q1.w;
    const float b2 = q2.x * vx + q2.y * vy + q2.w;
    const float b3 = q3.x * vx + q3.y * vy + q3.w;

    f32x4 rt, rw;
#pragma unroll
    for (int j = 0; j < kVpt; ++j) {
      const float vz = vz0 + (float)j * kStep;
      const float p0 = b0 + q0.z * vz;
      const float p1 = b1 + q1.z * vz;
      const float p2 = b2 + q2.z * vz;
      const float d  = b3 + q3.z * vz;          // voxel depth (cam z)

      const float pzd = p2 + 1e-8f;
      const float u = p0 / pzd;
      const float v = p1 / pzd;

      const bool valid_proj = (u >= 0.0f) & (u < (float)kW) &
                              (v >= 0.0f) & (v < (float)kH) & (d > 0.0f);

      // round-half-even, clamp, gather (always in-bounds after clamp)
      const float uf = fminf(fmaxf(rintf(u), 0.0f), (float)(kW - 1));
      const float vf = fminf(fmaxf(rintf(v), 0.0f), (float)(kH - 1));
      const int ui = (int)uf;
      const int vi = (int)vf;
      const float obs = depth[vi * kW + ui];

      const float sd    = obs - d;
      const float tval  = fminf(fmaxf(sd / kTrunc, -1.0f), 1.0f);
      const float wadd  = (valid_proj && (obs > 0.0f)) ? 1.0f : 0.0f;

      const float w_old = wv[j];
      const float t_old = tv[j];
      const float nw    = w_old + wadd;
      const float ut    = (nw > 0.0f) ? (w_old * t_old + wadd * tval) / nw : t_old;
      const float uw    = fminf(nw, kMaxW);
      rt[j] = ut;
      rw[j] = uw;
    }

    // streaming outputs: NT so the 134MB write stream doesn't evict the depth map
    __builtin_nontemporal_store(rt, (f32x4*)(tsdf_out + gbase));
    __builtin_nontemporal_store(rw, (f32x4*)(wght_out + gbase));
  }
}

// ------------------------------------------------------------------
extern "C" void kernel_launch(void* const* d_in, const int* in_sizes, int n_in,
                              void* d_out, int out_size, void* d_ws, size_t ws_size,
                              hipStream_t stream) {
  const float* depth = (const float*)d_in[0];   // [480*640]
  const float* pose  = (const float*)d_in[1];   // [16]
  const float* intr  = (const float*)d_in[2];   // [9]
  const float* tsdf  = (const float*)d_in[3];   // [256^3]
  const float* wght  = (const float*)d_in[4];   // [256^3]
  // d_in[5] (voxel_coords) intentionally unused: recomputed bit-exactly on device.
  (void)in_sizes; (void)n_in; (void)out_size; (void)ws_size;

  float* out_t = (float*)d_out;
  float* out_w = out_t + kNVox;
  float* Q     = (float*)d_ws;                  // 16 floats

  tsdf_prep_kernel<<<dim3(1), dim3(32), 0, stream>>>(pose, intr, Q);
  tsdf_fuse_kernel<<<dim3(kNBlk), dim3(kBlock), 0, stream>>>(
      depth, Q, tsdf, wght, out_t, out_w);
}